// InfiniteImpulseResponseDigitalFilter_31851477467886
// MI455X (gfx1250) — compile-verified
//
#include <hip/hip_runtime.h>

// ---------------------------------------------------------------------------
// IIR biquad over (512, 65536) fp32 via blocked linear-recurrence scan.
//   Pass 0: one-wave setup kernel -> per-lane WMMA constant table (4KB in ws)
//   Pass 1: per-chunk particular end-state, zero init      (WMMA engine)
//   Pass 2: tiny inter-chunk 2x2 affine scan per row
//   Pass 3: per-chunk outputs with correct init state      (WMMA engine)
// Memory-bound: >=256MB traffic @ 23.3TB/s ~ 11us floor; x (128MB) stays
// resident in the 192MB L2 between pass 1 and pass 3; y stored non-temporal.
//
// Segment math (16 timesteps x 16 independent columns per wave), FIR folded:
//   y[i] = sum_j w[i-j] x[t0+j] + r1[i] x[t0-1] + r2[i] x[t0-2]
//          + p[i] y[-1] + q[i] y[-2]
// with w = b (conv) h, r1[i]=b1 h[i]+b2 h[i-1], r2[i]=b2 h[i],
//      p[i]=h[i+1], q[i]=-a2 h[i].
// K extended to 20: A cols 0,1 = 0 (pad), col 2 = r2, col 3 = r1,
// cols 4..19 = Toeplitz(w).  B operand = raw x loaded directly in WMMA B
// layout (one b64 per K-group) => 5x V_WMMA_F32_16X16X4_F32 per 256 outputs.
// Constants are produced by the setup kernel and LOADED (not computed) by the
// hot kernels so the compiler cannot rematerialize them into the inner loop.
// ---------------------------------------------------------------------------

typedef float v8f __attribute__((ext_vector_type(8)));
typedef float v4f __attribute__((ext_vector_type(4)));
typedef float v2f __attribute__((ext_vector_type(2)));

constexpr int BATCH = 512;
constexpr int T     = 65536;
constexpr int L     = 1024;          // chunk length (2^10 -> 10 squarings)
constexpr int C     = T / L;         // 64 chunks per row
constexpr int SEGS  = L / 16;        // 64 WMMA segment steps per chunk
constexpr int COLS  = BATCH * C;     // 32768 independent sequences
constexpr int BLOCK = 256;           // 8 waves
constexpr int GRID  = (COLS / 16) / (BLOCK / 32);   // 2048 waves total
constexpr int CSTRIDE = 32;          // floats per lane in the constant table

// tap3(d) = t0*h[d] + t1*h[d-1] + t2*h[d-2], with h[0]=1,
// h[i] = -c1 h[i-1] - c2 h[i-2], h[<0]=0; returns 0 for d<0 (d<=16).
__device__ __forceinline__ float tap3(int d, float c1, float c2,
                                      float t0, float t1, float t2) {
  float him2 = 0.f, him1 = 0.f, hi = 1.f, r = 0.f;
#pragma unroll
  for (int i = 0; i <= 16; ++i) {
    if (i == d) r = t0 * hi + t1 * him1 + t2 * him2;
    float nxt = -c1 * hi - c2 * him1;
    him2 = him1; him1 = hi; hi = nxt;
  }
  return (d < 0) ? 0.f : r;
}

// Pass 0: per-lane constant table; identical for every wave of the hot
// kernels.  Layout per lane (base = lane*32 floats, 128B aligned):
//   [0..9]  A operand pairs (5 x v2f)   [12..19] pcf   [20..27] qcf
__global__ __launch_bounds__(32)
void iir_setup(const float* __restrict__ bc, const float* __restrict__ ac,
               float* __restrict__ coef)
{
  const int lane = threadIdx.x;
  const int half = lane >> 4;
  const int n    = lane & 15;

  const float a0 = ac[0];
  const float c1 = ac[1] / a0, c2 = ac[2] / a0;
  const float b0 = bc[0] / a0, b1 = bc[1] / a0, b2 = bc[2] / a0;

  float* out = coef + lane * CSTRIDE;
#pragma unroll
  for (int g = 0; g < 5; ++g) {
#pragma unroll
    for (int j = 0; j < 2; ++j) {
      const int k = 4 * g + 2 * half + j;
      float v;
      if (k < 2)       v = 0.f;
      else if (k == 2) v = tap3(n, c1, c2, b2, 0.f, 0.f);        // r2[m]
      else if (k == 3) v = tap3(n, c1, c2, b1, b2, 0.f);         // r1[m]
      else             v = tap3(n - k + 4, c1, c2, b0, b1, b2);  // w[m-(k-4)]
      out[2 * g + j] = v;
    }
  }
#pragma unroll
  for (int j = 0; j < 8; ++j) {
    const int M = j + 8 * half;
    out[12 + j] = tap3(M + 1, c1, c2, 1.f, 0.f, 0.f);       //  h[M+1]
    out[20 + j] = -c2 * tap3(M, c1, c2, 1.f, 0.f, 0.f);     // -c2 h[M]
  }
}

// One 16-timestep x 16-column segment step. ZERO_PAD only for the very first
// segment of a row (conv zero-padding + OOB avoidance at x[-4..-1]).
template <bool STORE_Y, bool ZERO_PAD>
__device__ __forceinline__
void segment_step(const float* __restrict__ x, float* __restrict__ y,
                  int t0, int half, int n, bool rowStart,
                  const v2f A[5], const float pcf[8], const float qcf[8],
                  float& yp1, float& yp2)
{
  const int tB = t0 - 4 + 2 * half;               // my B-operand load base

  v2f Bop[5];
#pragma unroll
  for (int g = 0; g < 5; ++g) {
    if (ZERO_PAD && g == 0 && rowStart) {
      Bop[0].x = 0.f; Bop[0].y = 0.f;
    } else {
      Bop[g] = *(const v2f*)(x + tB + 4 * g);     // 8B-aligned b64
    }
  }
  __builtin_prefetch(x + tB + 20, 0, 3);          // next segment's slice

  // D(16x16) = A(16x20) @ X(20x16): five K=4 WMMAs, f32 accumulate.
  v8f acc = {};
#pragma unroll
  for (int g = 0; g < 5; ++g)
    acc = __builtin_amdgcn_wmma_f32_16x16x4_f32(
        false, A[g], false, Bop[g], (short)0, acc, false, false);

  // Rank-2 recurrent-state update; element j <-> M = j + 8*half.
  float ynew[8];
#pragma unroll
  for (int j = 0; j < 8; ++j)
    ynew[j] = acc[j] + pcf[j] * yp1 + qcf[j] * yp2;

  // New state: y[15] (VGPR7) / y[14] (VGPR6) live in lanes 16-31.
  yp1 = __shfl(ynew[7], n + 16);
  yp2 = __shfl(ynew[6], n + 16);

  if constexpr (STORE_Y) {
    const int g0 = t0 + 8 * half;                 // my 8 contiguous outputs
    v4f o0 = {ynew[0], ynew[1], ynew[2], ynew[3]};
    v4f o1 = {ynew[4], ynew[5], ynew[6], ynew[7]};
    __builtin_nontemporal_store(o0, (v4f*)(y + g0));   // don't evict x from L2
    __builtin_nontemporal_store(o1, (v4f*)(y + g0 + 4));
  }
}

template <bool STORE_Y>
__global__ __launch_bounds__(BLOCK)
void iir_chunks(const float* __restrict__ x, const float* __restrict__ coef,
                float* __restrict__ y,
                float* __restrict__ stateIO)  // STORE_Y ? init-states(in) : end-states(out)
{
  const int lane = threadIdx.x & 31;
  const int wave = (blockIdx.x * blockDim.x + threadIdx.x) >> 5;
  const int half = lane >> 4;
  const int n    = lane & 15;

  const int wavesPerRow = C / 16;     // 4
  const int row    = wave / wavesPerRow;
  const int chunk0 = (wave % wavesPerRow) * 16;
  const int mycol  = chunk0 + n;                 // chunk index within row
  const int colBase = row * T + mycol * L;       // flat start index of my chunk

  // Load per-lane constants (produced by iir_setup) -- plain vector loads,
  // so they stay pinned in VGPRs instead of being rematerialized in-loop.
  const float* cb = coef + lane * CSTRIDE;
  const v4f ca0 = *(const v4f*)(cb + 0);
  const v4f ca1 = *(const v4f*)(cb + 4);
  const v2f ca2 = *(const v2f*)(cb + 8);
  const v2f A[5] = { {ca0.x, ca0.y}, {ca0.z, ca0.w},
                     {ca1.x, ca1.y}, {ca1.z, ca1.w},
                     {ca2.x, ca2.y} };
  const v4f cp0 = *(const v4f*)(cb + 12);
  const v4f cp1 = *(const v4f*)(cb + 16);
  const v4f cq0 = *(const v4f*)(cb + 20);
  const v4f cq1 = *(const v4f*)(cb + 24);
  const float pcf[8] = {cp0.x, cp0.y, cp0.z, cp0.w, cp1.x, cp1.y, cp1.z, cp1.w};
  const float qcf[8] = {cq0.x, cq0.y, cq0.z, cq0.w, cq1.x, cq1.y, cq1.z, cq1.w};

  // Chunk-initial recurrence state (y[-1], y[-2]).
  const int stateIdx = row * C + mycol;
  float yp1, yp2;
  if constexpr (STORE_Y) {
    yp1 = stateIO[2 * stateIdx + 0];
    yp2 = stateIO[2 * stateIdx + 1];
  } else {
    yp1 = 0.f; yp2 = 0.f;
  }

  const bool rowStart = (mycol == 0);

  // Peel segment 0 (the only one needing zero-padding); steady loop is clean.
  segment_step<STORE_Y, true>(x, y, colBase, half, n, rowStart,
                              A, pcf, qcf, yp1, yp2);
  for (int seg = 1; seg < SEGS; ++seg)
    segment_step<STORE_Y, false>(x, y, colBase + seg * 16, half, n, rowStart,
                                 A, pcf, qcf, yp1, yp2);

  if constexpr (!STORE_Y) {
    if (half == 0) {                    // one writer per column
      stateIO[2 * stateIdx + 0] = yp1;  // y[L-1]
      stateIO[2 * stateIdx + 1] = yp2;  // y[L-2]
    }
  }
}

// Pass 2: per-row affine scan over chunks: s_{c+1} = A^L s_c + e_c.
__global__ __launch_bounds__(256)
void iir_scan(const float* __restrict__ ac,
              const float* __restrict__ endSt, float* __restrict__ initSt)
{
  const int r = blockIdx.x * blockDim.x + threadIdx.x;
  if (r >= BATCH) return;
  const float a0 = ac[0], c1 = ac[1] / a0, c2 = ac[2] / a0;
  // M = A^L, A = [[-c1,-c2],[1,0]], L = 1024 = 2^10 -> 10 squarings.
  float m00 = -c1, m01 = -c2, m10 = 1.f, m11 = 0.f;
#pragma unroll
  for (int i = 0; i < 10; ++i) {
    const float n00 = m00 * m00 + m01 * m10;
    const float n01 = m00 * m01 + m01 * m11;
    const float n10 = m10 * m00 + m11 * m10;
    const float n11 = m10 * m01 + m11 * m11;
    m00 = n00; m01 = n01; m10 = n10; m11 = n11;
  }
  float s0 = 0.f, s1 = 0.f;                 // (y[-1], y[-2]) at row start
  for (int c = 0; c < C; ++c) {
    const int idx = r * C + c;
    initSt[2 * idx + 0] = s0;
    initSt[2 * idx + 1] = s1;
    const float e0 = endSt[2 * idx + 0];
    const float e1 = endSt[2 * idx + 1];
    const float t0 = m00 * s0 + m01 * s1 + e0;
    const float t1 = m10 * s0 + m11 * s1 + e1;
    s0 = t0; s1 = t1;
  }
}

extern "C" void kernel_launch(void* const* d_in, const int* in_sizes, int n_in,
                              void* d_out, int out_size, void* d_ws, size_t ws_size,
                              hipStream_t stream) {
  const float* x = (const float*)d_in[0];
  const float* b = (const float*)d_in[1];
  const float* a = (const float*)d_in[2];
  float* y = (float*)d_out;

  float* coefTab = (float*)d_ws;                  // 32*32 floats (4KB)
  float* endSt   = coefTab + 32 * CSTRIDE;        // 2 * 32768 floats
  float* initSt  = endSt + 2 * BATCH * C;         // 2 * 32768 floats

  iir_setup<<<1, 32, 0, stream>>>(b, a, coefTab);
  iir_chunks<false><<<GRID, BLOCK, 0, stream>>>(x, coefTab, nullptr, endSt);
  iir_scan<<<(BATCH + 255) / 256, 256, 0, stream>>>(a, endSt, initSt);
  iir_chunks<true><<<GRID, BLOCK, 0, stream>>>(x, coefTab, y, initSt);
}